// GATEdgeModel_23682449670193
// MI455X (gfx1250) — compile-verified
//
#include <hip/hip_runtime.h>
#include <math.h>

// ---------------------------------------------------------------------------
// GAT (2-layer, 4 heads) + mean pool + MLP, fp32 end-to-end.
// Matrix work uses V_WMMA_F32_16X16X4_F32 (exact fp32 on the CDNA5 matrix
// pipe); each wave computes a 16x64 strip (4 accumulators) to reuse the A
// fragment 4x and issue 4 back-to-back WMMAs per k-step.
// Edge softmax/scatter is L2-resident (Wh = 51MB < 192MB L2).
// ---------------------------------------------------------------------------

#define HEADS    4
#define HID      64
#define INF_DIM  128
#define HF       (HEADS * HID)      // 256
#define LRELU_A  0.2f
#define LN_EPS   1e-5f

typedef float v2f __attribute__((ext_vector_type(2)));
typedef float v8f __attribute__((ext_vector_type(8)));

// ---------------- degree / neighborhood feature ----------------------------

__global__ void k_deg(const int* __restrict__ col, float* __restrict__ deg, int E) {
    int e = blockIdx.x * blockDim.x + threadIdx.x;
    if (e < E) atomicAdd(&deg[col[e]], 1.0f);
}

__global__ void k_snd(const int* __restrict__ row, const float* __restrict__ deg,
                      float* __restrict__ snd, int E) {
    int e = blockIdx.x * blockDim.x + threadIdx.x;
    if (e < E) { int r = row[e]; atomicAdd(&snd[r], deg[r]); }
}

// ---------------- block reduction helper -----------------------------------

template <int BS>
__device__ __forceinline__ float block_sum(float v, float* sm) {
    int t = threadIdx.x;
    sm[t] = v; __syncthreads();
#pragma unroll
    for (int s = BS / 2; s > 0; s >>= 1) {
        if (t < s) sm[t] += sm[t + s];
        __syncthreads();
    }
    float r = sm[0]; __syncthreads();
    return r;
}

// node embedding (Linear(1,128)) + LayerNorm, one block (128 thr) per node
__global__ void k_embed_ln(const float* __restrict__ deg, const float* __restrict__ snd,
                           const float* __restrict__ W, const float* __restrict__ b,
                           const float* __restrict__ g, const float* __restrict__ beta,
                           float* __restrict__ x) {
    __shared__ float sm[INF_DIM];
    int n = blockIdx.x, f = threadIdx.x;
    float mnd = snd[n] / fmaxf(deg[n], 1.0f);
    float v = mnd * W[f] + b[f];
    float mu = block_sum<INF_DIM>(v, sm) * (1.0f / INF_DIM);
    float d  = v - mu;
    float var = block_sum<INF_DIM>(d * d, sm) * (1.0f / INF_DIM);
    x[(size_t)n * INF_DIM + f] = d * rsqrtf(var + LN_EPS) * g[f] + beta[f];
}

// ---------------- fp32 WMMA GEMM: C[M,Nc] = A[M,K] @ B[K,Nc] (+bias,relu) ---
// One 16x(16*TN) output strip per wave32 (TN accumulators, shared A frag);
// K stepped by 4 through v_wmma_f32_16x16x4_f32.
// Requires M%16==0, K%4==0, Nc%(16*TN)==0.
//
// Fragment layout assumptions (ISA 7.12.2, 16x16x4 f32):
//   A (16x4): lanes<->M, VGPR0 = K{0|2}, VGPR1 = K{1|3} (lo|hi lane half)
//   B (4x16): lanes<->N, VGPR0 = K{0|2}, VGPR1 = K{1|3}
//   C/D     : VGPR r holds M = r (lanes 0-15) / r+8 (lanes 16-31), N = lane&15

template <int TN>
__global__ void k_wmma_gemm(const float* __restrict__ A, const float* __restrict__ B,
                            const float* __restrict__ bias, float* __restrict__ C,
                            int M, int K, int Nc, int relu) {
    int strips_n = Nc / (16 * TN);
    int strip = blockIdx.x * (blockDim.x >> 5) + (threadIdx.x >> 5);
    int total = (M >> 4) * strips_n;
    if (strip >= total) return;
    int tm  = strip / strips_n;
    int tn0 = (strip % strips_n) * TN;
    int lane = threadIdx.x & 31;
    int half = lane >> 4;        // 0: K{0,1}, 1: K{2,3}
    int l15  = lane & 15;

    const float* arow = A + (size_t)(tm * 16 + l15) * K + 2 * half;   // lanes <-> M
    const float* bbase = B + (size_t)(tn0 * 16 + l15);                // lanes <-> N

    v8f acc[TN];
    const v8f vzero = {0.f, 0.f, 0.f, 0.f, 0.f, 0.f, 0.f, 0.f};
#pragma unroll
    for (int t = 0; t < TN; ++t) acc[t] = vzero;

    for (int k = 0; k < K; k += 4) {
        v2f af;
        af.x = arow[k];
        af.y = arow[k + 1];
        const float* brow0 = bbase + (size_t)(k + 2 * half) * Nc;
        const float* brow1 = brow0 + Nc;
#pragma unroll
        for (int t = 0; t < TN; ++t) {
            v2f bf;
            bf.x = brow0[t * 16];
            bf.y = brow1[t * 16];
            acc[t] = __builtin_amdgcn_wmma_f32_16x16x4_f32(
                /*neg_a=*/false, af, /*neg_b=*/false, bf,
                /*c_mod=*/(short)0, acc[t], /*reuse_a=*/false, /*reuse_b=*/false);
        }
    }
#pragma unroll
    for (int t = 0; t < TN; ++t) {
#pragma unroll
        for (int r = 0; r < 8; ++r) {
            int m = tm * 16 + r + 8 * half;
            int n = (tn0 + t) * 16 + l15;
            float v = acc[t][r];
            if (bias) v += bias[n];
            if (relu) v = fmaxf(v, 0.0f);
            C[(size_t)m * Nc + n] = v;
        }
    }
}

// ---------------- per-node attention logits es/ed --------------------------
// es[n,h] = Wh[n,h,:]·a[h,:64], ed[n,h] = Wh[n,h,:]·a[h,64:]

__global__ void k_node_attn(const float* __restrict__ Wh, const float* __restrict__ a,
                            float* __restrict__ es, float* __restrict__ ed, int N) {
    int i = blockIdx.x * blockDim.x + threadIdx.x;   // over N*HEADS
    if (i >= N * HEADS) return;
    int n = i >> 2, h = i & 3;
    const float* w  = Wh + (size_t)n * HF + h * HID;
    const float* as = a + h * (2 * HID);
    float s = 0.f, d = 0.f;
#pragma unroll 8
    for (int f = 0; f < HID; ++f) { s += w[f] * as[f]; d += w[f] * as[HID + f]; }
    es[i] = s; ed[i] = d;
}

// ---------------- ordered-uint float max encoding --------------------------

__device__ __forceinline__ unsigned enc_ord(float f) {
    unsigned u = __float_as_uint(f);
    return (u & 0x80000000u) ? ~u : (u | 0x80000000u);
}
__device__ __forceinline__ float dec_ord(unsigned u) {
    return __uint_as_float((u & 0x80000000u) ? (u ^ 0x80000000u) : ~u);
}

// pass 1: e = leakyrelu(es[src]+ed[dst]); segment max over dst
__global__ void k_edge_logits(const int* __restrict__ row, const int* __restrict__ col,
                              const float* __restrict__ es, const float* __restrict__ ed,
                              float* __restrict__ ebuf, unsigned* __restrict__ mbits,
                              int E, int N) {
    int e = blockIdx.x * blockDim.x + threadIdx.x;
    if (e >= E + N) return;
    int s = (e < E) ? row[e] : (e - E);
    int d = (e < E) ? col[e] : (e - E);
#pragma unroll
    for (int h = 0; h < HEADS; ++h) {
        float v = es[s * HEADS + h] + ed[d * HEADS + h];
        v = (v > 0.f) ? v : LRELU_A * v;
        ebuf[(size_t)e * HEADS + h] = v;
        atomicMax(&mbits[d * HEADS + h], enc_ord(v));
    }
}

// pass 2: ex = exp(e - max); segment sum of ex over dst
__global__ void k_edge_expsum(const int* __restrict__ col,
                              float* __restrict__ ebuf, const unsigned* __restrict__ mbits,
                              float* __restrict__ denom, int E, int N) {
    int e = blockIdx.x * blockDim.x + threadIdx.x;
    if (e >= E + N) return;
    int d = (e < E) ? col[e] : (e - E);
#pragma unroll
    for (int h = 0; h < HEADS; ++h) {
        float m  = dec_ord(mbits[d * HEADS + h]);
        float ex = expf(ebuf[(size_t)e * HEADS + h] - m);
        ebuf[(size_t)e * HEADS + h] = ex;
        atomicAdd(&denom[d * HEADS + h], ex);
    }
}

// pass 3: hp[dst] += alpha * Wh[src]; one block (64 thr = 2 waves) per edge,
// each thread handles 4 consecutive features: float4 gather (global_load_b128)
// + 4 consecutive global_atomic_add_f32.
__global__ void k_edge_scatter(const int* __restrict__ row, const int* __restrict__ col,
                               const float* __restrict__ ebuf, const float* __restrict__ denom,
                               const float* __restrict__ Wh, float* __restrict__ hp,
                               int E, int N) {
    int e = blockIdx.x;
    int j = threadIdx.x;                 // 0..63, features 4j..4j+3
    int s = (e < E) ? row[e] : (e - E);
    int d = (e < E) ? col[e] : (e - E);
    int h = j >> 4;                      // 16 threads per head (16*4 = 64 feats)
    float alpha = ebuf[(size_t)e * HEADS + h] / denom[d * HEADS + h];
    const float4 w = ((const float4*)(Wh + (size_t)s * HF))[j];
    float* out = hp + (size_t)d * HF + 4 * j;
    atomicAdd(out + 0, alpha * w.x);
    atomicAdd(out + 1, alpha * w.y);
    atomicAdd(out + 2, alpha * w.z);
    atomicAdd(out + 3, alpha * w.w);
}

// mean over heads + bias, LayerNorm, ReLU; one block (64 thr) per node
__global__ void k_finalize_h(const float* __restrict__ hp, const float* __restrict__ bias,
                             const float* __restrict__ g, const float* __restrict__ beta,
                             float* __restrict__ h) {
    __shared__ float sm[HID];
    int n = blockIdx.x, f = threadIdx.x;
    float v = 0.f;
#pragma unroll
    for (int hh = 0; hh < HEADS; ++hh)
        v += hp[(size_t)n * HF + hh * HID + f] + bias[hh * HID + f];
    v *= (1.0f / HEADS);
    float mu  = block_sum<HID>(v, sm) * (1.0f / HID);
    float d   = v - mu;
    float var = block_sum<HID>(d * d, sm) * (1.0f / HID);
    float o = d * rsqrtf(var + LN_EPS) * g[f] + beta[f];
    h[(size_t)n * HID + f] = fmaxf(o, 0.f);
}

// ---------------- pooling ---------------------------------------------------

__global__ void k_pool(const int* __restrict__ batch, const float* __restrict__ h1,
                       const float* __restrict__ h2, float* __restrict__ pooled,
                       float* __restrict__ cnt) {
    int n = blockIdx.x, f = threadIdx.x;     // 128 threads
    int b = batch[n];
    float v = (f < HID) ? h1[(size_t)n * HID + f] : h2[(size_t)n * HID + (f - HID)];
    atomicAdd(&pooled[b * (2 * HID) + f], v);
    if (f == 0) atomicAdd(&cnt[b], 1.0f);
}

__global__ void k_pool_div(float* __restrict__ pooled, const float* __restrict__ cnt, int G) {
    int i = blockIdx.x * blockDim.x + threadIdx.x;
    if (i < G * 2 * HID) pooled[i] /= fmaxf(cnt[i >> 7], 1.0f);
}

// ---------------------------------------------------------------------------

static inline void launch_gemm(const float* A, const float* B, const float* bias, float* C,
                               int M, int K, int Nc, int relu, hipStream_t stream) {
    if (Nc % 64 == 0) {
        int strips = (M / 16) * (Nc / 64);
        k_wmma_gemm<4><<<(strips + 7) / 8, 256, 0, stream>>>(A, B, bias, C, M, K, Nc, relu);
    } else {
        int strips = (M / 16) * (Nc / 16);
        k_wmma_gemm<1><<<(strips + 7) / 8, 256, 0, stream>>>(A, B, bias, C, M, K, Nc, relu);
    }
}

extern "C" void kernel_launch(void* const* d_in, const int* in_sizes, int n_in,
                              void* d_out, int out_size, void* d_ws, size_t ws_size,
                              hipStream_t stream) {
    (void)n_in; (void)ws_size;
    const int* edge_index = (const int*)d_in[0];       // [2,E]
    const int* batch      = (const int*)d_in[1];       // [N]
    const float* emb_W    = (const float*)d_in[2];
    const float* emb_b    = (const float*)d_in[3];
    const float* ln0_g    = (const float*)d_in[4];
    const float* ln0_b    = (const float*)d_in[5];
    const float* gat1_W   = (const float*)d_in[6];
    const float* gat1_a   = (const float*)d_in[7];
    const float* gat1_bias= (const float*)d_in[8];
    const float* ln1_g    = (const float*)d_in[9];
    const float* ln1_b    = (const float*)d_in[10];
    const float* gat2_W   = (const float*)d_in[11];
    const float* gat2_a   = (const float*)d_in[12];
    const float* gat2_bias= (const float*)d_in[13];
    const float* ln2_g    = (const float*)d_in[14];
    const float* ln2_b    = (const float*)d_in[15];
    const float* mlp1_W   = (const float*)d_in[16];
    const float* mlp1_b   = (const float*)d_in[17];
    const float* mlp2_W   = (const float*)d_in[18];
    const float* mlp2_b   = (const float*)d_in[19];
    const float* mlp3_W   = (const float*)d_in[20];
    const float* mlp3_b   = (const float*)d_in[21];

    const int E  = in_sizes[0] / 2;
    const int N  = in_sizes[1];
    const int G  = out_size / 16;      // OUT = 16
    const int E2 = E + N;
    const int* row = edge_index;
    const int* col = edge_index + E;

    // ---- carve workspace -------------------------------------------------
    size_t off = 0;
    auto carve = [&](size_t bytes) -> void* {
        off = (off + 255) & ~(size_t)255;
        void* p = (char*)d_ws + off;
        off += bytes;
        return p;
    };
    float*    deg    = (float*)carve((size_t)N * sizeof(float));
    float*    snd    = (float*)carve((size_t)N * sizeof(float));
    float*    x      = (float*)carve((size_t)N * INF_DIM * sizeof(float));
    float*    Wh     = (float*)carve((size_t)N * HF * sizeof(float));   // reused layer 1/2
    float*    hp     = (float*)carve((size_t)N * HF * sizeof(float));   // reused layer 1/2
    float*    es     = (float*)carve((size_t)N * HEADS * sizeof(float));
    float*    ed     = (float*)carve((size_t)N * HEADS * sizeof(float));
    unsigned* mbits  = (unsigned*)carve((size_t)N * HEADS * sizeof(unsigned));
    float*    denom  = (float*)carve((size_t)N * HEADS * sizeof(float));
    float*    ebuf   = (float*)carve((size_t)E2 * HEADS * sizeof(float));
    float*    h1     = (float*)carve((size_t)N * HID * sizeof(float));
    float*    h2     = (float*)carve((size_t)N * HID * sizeof(float));
    float*    pooled = (float*)carve((size_t)G * 2 * HID * sizeof(float));
    float*    cnt    = (float*)carve((size_t)G * sizeof(float));
    float*    z1     = (float*)carve((size_t)G * 2 * HID * sizeof(float));
    float*    z2     = (float*)carve((size_t)G * HID * sizeof(float));

    const int TB = 256;
    const int gE  = (E  + TB - 1) / TB;
    const int gE2 = (E2 + TB - 1) / TB;

    // ---- neighborhood feature + embedding + LN0 --------------------------
    hipMemsetAsync(deg, 0, (size_t)N * sizeof(float), stream);
    hipMemsetAsync(snd, 0, (size_t)N * sizeof(float), stream);
    k_deg<<<gE, TB, 0, stream>>>(col, deg, E);
    k_snd<<<gE, TB, 0, stream>>>(row, deg, snd, E);
    k_embed_ln<<<N, INF_DIM, 0, stream>>>(deg, snd, emb_W, emb_b, ln0_g, ln0_b, x);

    // ---- GAT layer 1 -----------------------------------------------------
    launch_gemm(x, gat1_W, nullptr, Wh, N, INF_DIM, HF, 0, stream);
    k_node_attn<<<(N * HEADS + TB - 1) / TB, TB, 0, stream>>>(Wh, gat1_a, es, ed, N);
    hipMemsetAsync(mbits, 0, (size_t)N * HEADS * sizeof(unsigned), stream);
    hipMemsetAsync(denom, 0, (size_t)N * HEADS * sizeof(float), stream);
    hipMemsetAsync(hp,    0, (size_t)N * HF * sizeof(float), stream);
    k_edge_logits <<<gE2, TB, 0, stream>>>(row, col, es, ed, ebuf, mbits, E, N);
    k_edge_expsum <<<gE2, TB, 0, stream>>>(col, ebuf, mbits, denom, E, N);
    k_edge_scatter<<<E2, HID, 0, stream>>>(row, col, ebuf, denom, Wh, hp, E, N);
    k_finalize_h  <<<N, HID, 0, stream>>>(hp, gat1_bias, ln1_g, ln1_b, h1);

    // ---- GAT layer 2 -----------------------------------------------------
    launch_gemm(h1, gat2_W, nullptr, Wh, N, HID, HF, 0, stream);
    k_node_attn<<<(N * HEADS + TB - 1) / TB, TB, 0, stream>>>(Wh, gat2_a, es, ed, N);
    hipMemsetAsync(mbits, 0, (size_t)N * HEADS * sizeof(unsigned), stream);
    hipMemsetAsync(denom, 0, (size_t)N * HEADS * sizeof(float), stream);
    hipMemsetAsync(hp,    0, (size_t)N * HF * sizeof(float), stream);
    k_edge_logits <<<gE2, TB, 0, stream>>>(row, col, es, ed, ebuf, mbits, E, N);
    k_edge_expsum <<<gE2, TB, 0, stream>>>(col, ebuf, mbits, denom, E, N);
    k_edge_scatter<<<E2, HID, 0, stream>>>(row, col, ebuf, denom, Wh, hp, E, N);
    k_finalize_h  <<<N, HID, 0, stream>>>(hp, gat2_bias, ln2_g, ln2_b, h2);

    // ---- JumpingKnowledge cat + global mean pool -------------------------
    hipMemsetAsync(pooled, 0, (size_t)G * 2 * HID * sizeof(float), stream);
    hipMemsetAsync(cnt,    0, (size_t)G * sizeof(float), stream);
    k_pool    <<<N, 2 * HID, 0, stream>>>(batch, h1, h2, pooled, cnt);
    k_pool_div<<<(G * 2 * HID + TB - 1) / TB, TB, 0, stream>>>(pooled, cnt, G);

    // ---- MLP head (WMMA, fused bias/relu) --------------------------------
    launch_gemm(pooled, mlp1_W, mlp1_b, z1, G, 2 * HID, 2 * HID, 1, stream);
    launch_gemm(z1,     mlp2_W, mlp2_b, z2, G, 2 * HID, HID,     1, stream);
    launch_gemm(z2,     mlp3_W, mlp3_b, (float*)d_out, G, HID, 16, 0, stream);
}